// HashEncoder_89790586290649
// MI455X (gfx1250) — compile-verified
//
#include <hip/hip_runtime.h>
#include <stdint.h>

// ---------------------------------------------------------------------------
// Instant-NGP style multiresolution hash-grid encode for MI455X (gfx1250).
//   x:      [N,3]  f32   (N = 1048576)
//   tables: [L=16, T=16384, F=2] f32  (128 KB per level, 2 MB total)
//   out:    [N, 32] f32  (level i -> columns 2i, 2i+1)
//
// Strategy: block = (point chunk, level PAIR). Two level tables = 256 KB are
// staged into LDS with CDNA5 async global->LDS copies (ASYNCcnt), then all
// gathers are ds_load_b64 out of LDS. Each thread emits a contiguous float4
// (two levels' features) per point -> coalesced b128 stores.
// ---------------------------------------------------------------------------

#define HG_L         16
#define HG_T         16384
#define HG_TMASK     16383u
#define HG_BLOCK     512
#define HG_CHUNK     8192
#define HG_PAIR_BYTES (2 * HG_T * 2 * 4)   // 262144 bytes = two level tables

// n_l / 2 for n_l = floor(16 * 2^(i/3)):
// {16,20,25,32,40,50,64,80,101,128,161,203,256,322,406,512}
__constant__ float kHalfRes[HG_L] = {
    8.0f, 10.0f, 12.5f, 16.0f, 20.0f, 25.0f, 32.0f, 40.0f,
    50.5f, 64.0f, 80.5f, 101.5f, 128.0f, 161.0f, 203.0f, 256.0f
};

__device__ __forceinline__ void encode_level(
    const float2* __restrict__ tab,   // LDS-resident table for this level
    float px, float py, float pz, float s,
    float& o0, float& o1)
{
    // rel = (x - (-1)) / (2/n_l) = x*(n_l/2) + (n_l/2)
    float rx = fmaf(px, s, s);
    float ry = fmaf(py, s, s);
    float rz = fmaf(pz, s, s);
    float bx = floorf(rx), by = floorf(ry), bz = floorf(rz);
    float fx = rx - bx,    fy = ry - by,    fz = rz - bz;

    uint32_t ix = (uint32_t)bx, iy = (uint32_t)by, iz = (uint32_t)bz;
    uint32_t hx0 = ix;               uint32_t hx1 = ix + 1u;
    uint32_t hy0 = iy * 2654435761u; uint32_t hy1 = hy0 + 2654435761u;
    uint32_t hz0 = iz * 805459861u;  uint32_t hz1 = hz0 + 805459861u;

    // 8 LDS gathers (ds_load_b64), T power-of-two so %T == &TMASK
    float2 v000 = tab[(hx0 ^ hy0 ^ hz0) & HG_TMASK];
    float2 v001 = tab[(hx0 ^ hy0 ^ hz1) & HG_TMASK];
    float2 v010 = tab[(hx0 ^ hy1 ^ hz0) & HG_TMASK];
    float2 v011 = tab[(hx0 ^ hy1 ^ hz1) & HG_TMASK];
    float2 v100 = tab[(hx1 ^ hy0 ^ hz0) & HG_TMASK];
    float2 v101 = tab[(hx1 ^ hy0 ^ hz1) & HG_TMASK];
    float2 v110 = tab[(hx1 ^ hy1 ^ hz0) & HG_TMASK];
    float2 v111 = tab[(hx1 ^ hy1 ^ hz1) & HG_TMASK];

    float gx = 1.0f - fx, gy = 1.0f - fy, gz = 1.0f - fz;
    // ((wx*wy)*wz) association, corner order (0,0,0),(0,0,1),...,(1,1,1)
    float q00 = gx * gy, q01 = gx * fy, q10 = fx * gy, q11 = fx * fy;
    float w000 = q00 * gz, w001 = q00 * fz;
    float w010 = q01 * gz, w011 = q01 * fz;
    float w100 = q10 * gz, w101 = q10 * fz;
    float w110 = q11 * gz, w111 = q11 * fz;

    float a0 = w000 * v000.x;            float a1 = w000 * v000.y;
    a0 = fmaf(w001, v001.x, a0);         a1 = fmaf(w001, v001.y, a1);
    a0 = fmaf(w010, v010.x, a0);         a1 = fmaf(w010, v010.y, a1);
    a0 = fmaf(w011, v011.x, a0);         a1 = fmaf(w011, v011.y, a1);
    a0 = fmaf(w100, v100.x, a0);         a1 = fmaf(w100, v100.y, a1);
    a0 = fmaf(w101, v101.x, a0);         a1 = fmaf(w101, v101.y, a1);
    a0 = fmaf(w110, v110.x, a0);         a1 = fmaf(w110, v110.y, a1);
    a0 = fmaf(w111, v111.x, a0);         a1 = fmaf(w111, v111.y, a1);
    o0 = a0; o1 = a1;
}

extern "C" __global__ __launch_bounds__(HG_BLOCK, 1)
void hashgrid_pair_kernel(const float* __restrict__ x,
                          const float* __restrict__ tables,
                          float* __restrict__ out,
                          int n)
{
    extern __shared__ float smem[];          // 256 KB: levels 2p and 2p+1
    const int pair = (int)blockIdx.y;        // 0..7

    // ------- stage two level tables into LDS with async copies -------------
    {
        const char* src = (const char*)tables + (size_t)pair * HG_PAIR_BYTES;
        // flat LDS address: high 32 bits = shared aperture, low 32 = offset
        uint32_t lds_base = (uint32_t)(size_t)(void*)smem;
        #pragma unroll
        for (int it = 0; it < HG_PAIR_BYTES / (HG_BLOCK * 16); ++it) {
            uint32_t off = (uint32_t)((it * HG_BLOCK + (int)threadIdx.x) * 16);
            uint32_t dst = lds_base + off;
            // CDNA5 async global->LDS, 16B per lane, tracked by ASYNCcnt
            asm volatile("global_load_async_to_lds_b128 %0, %1, %2"
                         :: "v"(dst), "v"(off), "s"(src)
                         : "memory");
        }
        asm volatile("s_wait_asynccnt 0x0" ::: "memory");
    }
    __syncthreads();

    const float2* tabA = reinterpret_cast<const float2*>(smem);          // level 2p
    const float2* tabB = tabA + HG_T;                                    // level 2p+1
    const float sA = kHalfRes[2 * pair];
    const float sB = kHalfRes[2 * pair + 1];

    const int base = (int)blockIdx.x * HG_CHUNK;
    int end = base + HG_CHUNK;
    if (end > n) end = n;

    for (int p = base + (int)threadIdx.x; p < end; p += HG_BLOCK) {
        const float px = x[3 * p + 0];
        const float py = x[3 * p + 1];
        const float pz = x[3 * p + 2];

        float a0, a1, b0, b1;
        encode_level(tabA, px, py, pz, sA, a0, a1);
        encode_level(tabB, px, py, pz, sB, b0, b1);

        // columns [4p .. 4p+3] of the [N,32] output -> one b128 store
        *reinterpret_cast<float4*>(out + (size_t)p * 32 + 4 * pair) =
            make_float4(a0, a1, b0, b1);
    }
}

extern "C" void kernel_launch(void* const* d_in, const int* in_sizes, int n_in,
                              void* d_out, int out_size, void* d_ws, size_t ws_size,
                              hipStream_t stream) {
    (void)n_in; (void)out_size; (void)d_ws; (void)ws_size;
    const float* x      = (const float*)d_in[0];
    const float* tables = (const float*)d_in[1];
    float* out          = (float*)d_out;

    const int n = in_sizes[0] / 3;
    if (n <= 0) return;

    // Opt in to >64KB dynamic LDS (capture-safe, deterministic, idempotent).
    (void)hipFuncSetAttribute((const void*)hashgrid_pair_kernel,
                              hipFuncAttributeMaxDynamicSharedMemorySize,
                              HG_PAIR_BYTES);

    dim3 grid((n + HG_CHUNK - 1) / HG_CHUNK, HG_L / 2);
    hashgrid_pair_kernel<<<grid, HG_BLOCK, HG_PAIR_BYTES, stream>>>(
        x, tables, out, n);
}